// Sparse_MatMul_55937654063758
// MI455X (gfx1250) — compile-verified
//
#include <hip/hip_runtime.h>
#include <stdint.h>

typedef __attribute__((ext_vector_type(2))) float v2f;
typedef __attribute__((ext_vector_type(8))) float v8f;

#define F_DIM 256
#define CHUNK 512   // edges staged in LDS per iteration (4 KB)

// ---------------- Phase 0: zero counters ----------------
__global__ void zero_kernel(int* __restrict__ p, int n) {
    int i = blockIdx.x * blockDim.x + threadIdx.x;
    if (i < n) p[i] = 0;
}

// ---------------- Phase 1: histogram by 16-row block (LDS sub-hist) ----------------
__global__ void hist_kernel(const int* __restrict__ row, int E,
                            int* __restrict__ counts, int nb) {
    extern __shared__ int lh[];
    for (int i = threadIdx.x; i < nb; i += blockDim.x) lh[i] = 0;
    __syncthreads();
    int stride = gridDim.x * blockDim.x;
    for (int i = blockIdx.x * blockDim.x + threadIdx.x; i < E; i += stride)
        atomicAdd(&lh[row[i] >> 4], 1);
    __syncthreads();
    for (int i = threadIdx.x; i < nb; i += blockDim.x) {
        int v = lh[i];
        if (v) atomicAdd(&counts[i], v);
    }
}

// ---------------- Phase 2: exclusive scan (single workgroup) ----------------
__global__ void scan_kernel(const int* __restrict__ counts, int nb,
                            int* __restrict__ offsets) {
    __shared__ int lsum[1024];
    int t = threadIdx.x;
    int chunk = (nb + 1023) >> 10;
    int beg = t * chunk;
    int end = beg + chunk; if (end > nb) end = nb;
    int s = 0;
    for (int i = beg; i < end && i < nb; ++i) s += counts[i];
    lsum[t] = s;
    __syncthreads();
    for (int d = 1; d < 1024; d <<= 1) {
        int v = (t >= d) ? lsum[t - d] : 0;
        __syncthreads();
        lsum[t] += v;
        __syncthreads();
    }
    int run = (t == 0) ? 0 : lsum[t - 1];
    for (int i = beg; i < end && i < nb; ++i) { offsets[i] = run; run += counts[i]; }
}

// ---------------- Phase 3: scatter edges into bins ----------------
__global__ void scatter_kernel(const int* __restrict__ row, const int* __restrict__ col,
                               const float* __restrict__ val, int E,
                               const int* __restrict__ offsets, int* __restrict__ cursor,
                               float* __restrict__ bval, unsigned* __restrict__ bcol) {
    int i = blockIdx.x * blockDim.x + threadIdx.x;
    if (i < E) {
        int r = row[i];
        int b = r >> 4;
        int pos = offsets[b] + atomicAdd(&cursor[b], 1);
        bval[pos] = val[i];
        bcol[pos] = (unsigned)col[i] | ((unsigned)(r & 15) << 27);
    }
}

// ---------------- Phase 4: WMMA SpMM ----------------
// One 256-thread workgroup (8 wave32) per 16-row output block.
// Wave w owns features [32w, 32w+32) as two 16-wide WMMA tiles (c0, c1),
// so the A fragment + edge metadata are amortized over 2 WMMAs.
// Edge bin is staged through LDS in 512-edge chunks, zero-padded to a
// multiple of 4 so the inner loop has no bounds selects (EXEC all-1s at WMMA).
//   A(16x4) f32 : A[m][k] = (edge k local-row == m) ? value_k : 0
//                 lanes 0-15 -> M; v0 = K0/K2 (by lane half), v1 = K1/K3
//   B(4x16) f32 : B[k][n]  = mat2[col_k][featbase+n]; lanes -> N, VGPR/half -> K
//   C += A*B via v_wmma_f32_16x16x4_f32 (full f32 precision)
__global__ void __launch_bounds__(256)
spmm_wmma_kernel(const float* __restrict__ bval, const unsigned* __restrict__ bcol,
                 const int* __restrict__ offsets, const int* __restrict__ counts,
                 const float* __restrict__ mat2, float* __restrict__ out) {
    __shared__ float    sval[CHUNK];
    __shared__ unsigned scol[CHUNK];

    const int blk  = blockIdx.x;
    const int lane = threadIdx.x & 31;
    const int wave = threadIdx.x >> 5;
    const int half = lane >> 4;      // 0: K=0,1 ; 1: K=2,3
    const int ln   = lane & 15;      // M index for A, N index for B
    const int fb0  = wave * 32;
    const int fb1  = fb0 + 16;

    const int start = offsets[blk];
    const int cnt   = counts[blk];

    v8f c0 = {};
    v8f c1 = {};

    for (int cbase = 0; cbase < cnt; cbase += CHUNK) {
        int ccnt = cnt - cbase; if (ccnt > CHUNK) ccnt = CHUNK;
        int padded = (ccnt + 3) & ~3;          // zero-pad to multiple of 4

        for (int i = threadIdx.x; i < padded; i += 256) {
            if (i < ccnt) {
                sval[i] = bval[start + cbase + i];
                scol[i] = bcol[start + cbase + i];
            } else {
                sval[i] = 0.0f;                // padded edge: value 0, col 0
                scol[i] = 0u;
            }
        }
        __syncthreads();

        const int ngroups = padded >> 2;
        for (int g = 0; g < ngroups; ++g) {
            int base = g << 2;
            int eA = base + (half << 1);
            int eB = eA + 1;

            float    vA = sval[eA];
            float    vB = sval[eB];
            unsigned pA = scol[eA];
            unsigned pB = scol[eB];

            int colA = (int)(pA & 0x07FFFFFFu);
            int colB = (int)(pB & 0x07FFFFFFu);
            int mrA  = (int)(pA >> 27);
            int mrB  = (int)(pB >> 27);

            v2f a;
            a.x = (mrA == ln) ? vA : 0.0f;
            a.y = (mrB == ln) ? vB : 0.0f;

            const float* rA = mat2 + (size_t)colA * F_DIM;
            const float* rB = mat2 + (size_t)colB * F_DIM;

            v2f b0; b0.x = rA[fb0 + ln]; b0.y = rB[fb0 + ln];
            v2f b1; b1.x = rA[fb1 + ln]; b1.y = rB[fb1 + ln];

            // 8 args: (neg_a, A, neg_b, B, c_mod, C, reuse_a, reuse_b)
            c0 = __builtin_amdgcn_wmma_f32_16x16x4_f32(
                     false, a, false, b0, (short)0, c0, false, false);
            c1 = __builtin_amdgcn_wmma_f32_16x16x4_f32(
                     false, a, false, b1, (short)0, c1, false, false);
        }
        __syncthreads();
    }

    // C/D layout: VGPR j -> M=j (lanes 0-15) / M=j+8 (lanes 16-31); lanes -> N
    const int rowbase = blk * 16 + half * 8;
    float* o0 = out + (size_t)rowbase * F_DIM + fb0 + ln;
    float* o1 = out + (size_t)rowbase * F_DIM + fb1 + ln;
#pragma unroll
    for (int j = 0; j < 8; ++j) {
        o0[(size_t)j * F_DIM] = c0[j];
        o1[(size_t)j * F_DIM] = c1[j];
    }
}

// ---------------- host-side launcher ----------------
extern "C" void kernel_launch(void* const* d_in, const int* in_sizes, int n_in,
                              void* d_out, int out_size, void* d_ws, size_t ws_size,
                              hipStream_t stream) {
    const int*   indices = (const int*)d_in[0];   // [2, E] int32
    const float* values  = (const float*)d_in[1]; // [E] f32
    const float* mat2    = (const float*)d_in[3]; // [N, 256] f32

    const int E  = in_sizes[1];
    const int N  = in_sizes[3] / F_DIM;
    const int nb = (N + 15) >> 4;                 // 16-row blocks

    const int* row = indices;
    const int* col = indices + E;

    // Workspace layout: counts | offsets | cursor | bval | bcol
    char* ws = (char*)d_ws;
    int* counts  = (int*)ws;  ws += (size_t)nb * sizeof(int);
    int* offsets = (int*)ws;  ws += (size_t)nb * sizeof(int);
    int* cursor  = (int*)ws;  ws += (size_t)nb * sizeof(int);
    uintptr_t u = ((uintptr_t)ws + 15) & ~(uintptr_t)15;
    ws = (char*)u;
    float*    bval = (float*)ws;    ws += (size_t)E * sizeof(float);
    unsigned* bcol = (unsigned*)ws;

    zero_kernel<<<(3 * nb + 255) / 256, 256, 0, stream>>>(counts, 3 * nb);
    hist_kernel<<<512, 256, nb * sizeof(int), stream>>>(row, E, counts, nb);
    scan_kernel<<<1, 1024, 0, stream>>>(counts, nb, offsets);
    scatter_kernel<<<(E + 255) / 256, 256, 0, stream>>>(row, col, values, E,
                                                        offsets, cursor, bval, bcol);
    spmm_wmma_kernel<<<nb, 256, 0, stream>>>(bval, bcol, offsets, counts, mat2,
                                             (float*)d_out);
}